// MambaForecaster_60722247631082
// MI455X (gfx1250) — compile-verified
//
#include <hip/hip_runtime.h>

typedef __attribute__((ext_vector_type(2))) float v2f;
typedef __attribute__((ext_vector_type(8))) float v8f;

#define B_   8
#define L_   2048
#define DM_  1024
#define DI_  2048
#define DS_  16
#define DTR_ 64
#define NX_  96   /* DTR + 2*DS */

static __device__ __forceinline__ float fast_sigmoid(float x) {
  return 1.0f / (1.0f + __expf(-x));
}

// ---------------------------------------------------------------------------
// k0: v[j] = sum_d w_in[d]*ipw[d,j] ; c[j] = sum_d b_in[d]*ipw[d,j]  (j<4096)
// ---------------------------------------------------------------------------
__global__ void k_vc(const float* __restrict__ w_in, const float* __restrict__ b_in,
                     const float* __restrict__ ipw,
                     float* __restrict__ v, float* __restrict__ c) {
  int j = blockIdx.x * blockDim.x + threadIdx.x;
  if (j >= 2 * DI_) return;
  float av = 0.f, ac = 0.f;
  for (int d = 0; d < DM_; ++d) {
    float w = ipw[(size_t)d * (2 * DI_) + j];
    av = fmaf(w_in[d], w, av);
    ac = fmaf(b_in[d], w, ac);
  }
  v[j] = av;
  c[j] = ac;
}

// ---------------------------------------------------------------------------
// k1: u[b,t,d] = silu( conv(x*v+c) + cb ), 4 channels per thread (float4 path)
// ---------------------------------------------------------------------------
__global__ void k_uconv(const float* __restrict__ x, const float4* __restrict__ v4,
                        const float4* __restrict__ c4, const float4* __restrict__ cw4,
                        const float4* __restrict__ cb4, float4* __restrict__ U4) {
  size_t qidx = (size_t)blockIdx.x * blockDim.x + threadIdx.x;  // quad index
  int dq = (int)(qidx & (DI_ / 4 - 1));      // 512 quads per row
  size_t row = qidx >> 9;                    // b*L + t
  int t = (int)(row & (L_ - 1));
  int b = (int)(row >> 11);
  float4 vv = v4[dq];
  float4 cc = c4[dq];
  float4 cb = cb4[dq];
  float4 w0 = cw4[dq * 4 + 0];   // channel 4dq+0 taps
  float4 w1 = cw4[dq * 4 + 1];
  float4 w2 = cw4[dq * 4 + 2];
  float4 w3 = cw4[dq * 4 + 3];
  const float* xb = x + (size_t)b * L_;
  float xv[4];
#pragma unroll
  for (int k = 0; k < 4; ++k) {
    int tau = t - 3 + k;
    xv[k] = (tau >= 0) ? xb[tau] : 0.f;
  }
  float msk[4];
#pragma unroll
  for (int k = 0; k < 4; ++k) msk[k] = (t - 3 + k >= 0) ? 1.f : 0.f;

  float acc[4] = {cb.x, cb.y, cb.z, cb.w};
  const float* vvp = (const float*)&vv;
  const float* ccp = (const float*)&cc;
  const float4* wrow[4] = {&w0, &w1, &w2, &w3};
#pragma unroll
  for (int j = 0; j < 4; ++j) {
    const float* w = (const float*)wrow[j];
#pragma unroll
    for (int k = 0; k < 4; ++k) {
      float term = fmaf(xv[k], vvp[j], ccp[j]) * msk[k];
      acc[j] = fmaf(term, w[k], acc[j]);
    }
  }
  float4 outv;
  outv.x = acc[0] * fast_sigmoid(acc[0]);
  outv.y = acc[1] * fast_sigmoid(acc[1]);
  outv.z = acc[2] * fast_sigmoid(acc[2]);
  outv.w = acc[3] * fast_sigmoid(acc[3]);
  U4[qidx] = outv;
}

// ---------------------------------------------------------------------------
// k2: x_dbl = U (16384 x 2048) @ x_proj_w (2048 x 96)  — WMMA f32 16x16x4
// ---------------------------------------------------------------------------
__global__ void k_gemmA(const float* __restrict__ U, const float* __restrict__ P,
                        float* __restrict__ X) {
  const int lane = threadIdx.x & 31;
  const int wave = threadIdx.x >> 5;
  const int m0 = (blockIdx.x * 8 + wave) * 16;
  const int lo = lane & 15;
  const int hi = lane >> 4;
  v8f zero = {};
  v8f acc[6];
#pragma unroll
  for (int j = 0; j < 6; ++j) acc[j] = zero;
  const float* arow = U + (size_t)(m0 + lo) * DI_;
  for (int k = 0; k < DI_; k += 4) {
    int ka = k + 2 * hi;
    v2f a;
    a.x = arow[ka];
    a.y = arow[ka + 1];
#pragma unroll
    for (int j = 0; j < 6; ++j) {
      v2f b;
      b.x = P[(size_t)ka * NX_ + j * 16 + lo];
      b.y = P[(size_t)(ka + 1) * NX_ + j * 16 + lo];
      acc[j] = __builtin_amdgcn_wmma_f32_16x16x4_f32(
          false, a, false, b, (short)0, acc[j], false, false);
    }
  }
#pragma unroll
  for (int r = 0; r < 8; ++r) {
    int rowm = m0 + r + 8 * hi;
#pragma unroll
    for (int j = 0; j < 6; ++j)
      X[(size_t)rowm * NX_ + j * 16 + lo] = acc[j][r];
  }
}

// ---------------------------------------------------------------------------
// k3: delta = softplus( x_dbl[:, :64] @ dt_proj_w (64x2048) + b ) — WMMA
// ---------------------------------------------------------------------------
__global__ void k_gemmB(const float* __restrict__ XD, const float* __restrict__ W,
                        const float* __restrict__ bias, float* __restrict__ DEL) {
  const int lane = threadIdx.x & 31;
  const int wave = threadIdx.x >> 5;
  const int m0 = blockIdx.x * 16;
  const int lo = lane & 15;
  const int hi = lane >> 4;
  v2f afrag[16];
  const float* arow = XD + (size_t)(m0 + lo) * NX_;
#pragma unroll
  for (int ks = 0; ks < 16; ++ks) {
    int ka = ks * 4 + 2 * hi;
    afrag[ks].x = arow[ka];
    afrag[ks].y = arow[ka + 1];
  }
  for (int nt = 0; nt < 16; ++nt) {
    int n0 = (wave * 16 + nt) * 16;
    v8f acc = {};
#pragma unroll
    for (int ks = 0; ks < 16; ++ks) {
      int ka = ks * 4 + 2 * hi;
      v2f b;
      b.x = W[(size_t)ka * DI_ + n0 + lo];
      b.y = W[(size_t)(ka + 1) * DI_ + n0 + lo];
      acc = __builtin_amdgcn_wmma_f32_16x16x4_f32(
          false, afrag[ks], false, b, (short)0, acc, false, false);
    }
    float bv = bias[n0 + lo];
#pragma unroll
    for (int r = 0; r < 8; ++r) {
      float t = acc[r] + bv;
      float sp = (t > 20.f) ? t : log1pf(__expf(t));
      DEL[(size_t)(m0 + r + 8 * hi) * DI_ + n0 + lo] = sp;
    }
  }
}

// ---------------------------------------------------------------------------
// k4: selective scan, 2 threads per channel (8 states each) -> 1024 waves.
// Block: 128 channels x 2 state-halves. B_t staged in LDS per 64-step chunk.
// Next chunk's delta/u rows prefetched while processing current chunk.
// ---------------------------------------------------------------------------
__global__ void k_scan(const float* __restrict__ DEL, const float* __restrict__ U,
                       const float* __restrict__ XD, const float* __restrict__ A_log,
                       const float* __restrict__ Dp, const float* __restrict__ x,
                       const float* __restrict__ v, const float* __restrict__ c,
                       float* __restrict__ YG) {
  __shared__ __align__(16) float Bs[64 * DS_];
  __shared__ float yred[256];
  const int tid = threadIdx.x;
  const int b = blockIdx.x >> 4;                  // 16 blocks per batch
  const int d = ((blockIdx.x & 15) << 7) + (tid & 127);
  const int shalf = tid >> 7;                     // 0 or 1 (wave-uniform)
  const int base = shalf << 3;                    // state offset 0 or 8
  const float LOG2E = 1.4426950408889634f;

  float as2[8];
#pragma unroll
  for (int j = 0; j < 8; ++j)
    as2[j] = -__expf(A_log[(size_t)d * DS_ + base + j]) * LOG2E;

  // Fast path when A[d,s] proportional to (s+1) (true for this model).
  const float g_unit = as2[0] / (float)(base + 1);
  bool fast = true;
#pragma unroll
  for (int j = 1; j < 8; ++j) {
    float pred = g_unit * (float)(base + j + 1);
    if (fabsf(as2[j] - pred) > 1e-4f * fabsf(as2[j]) + 1e-6f) fast = false;
  }

  float h[8];
#pragma unroll
  for (int j = 0; j < 8; ++j) h[j] = 0.f;

  const size_t rowbase = (size_t)b * L_;
  float u_v = 0.f;

  for (int t0 = 0; t0 < L_; t0 += 64) {
    for (int i = tid; i < 64 * DS_; i += 256) {
      int tt = i >> 4, s = i & 15;
      Bs[i] = XD[(rowbase + t0 + tt) * NX_ + DTR_ + s];
    }
    __syncthreads();
    const bool do_pf = (t0 + 64 < L_);
    if (fast) {
      for (int tt = 0; tt < 64; ++tt) {
        const size_t row = rowbase + t0 + tt;
        const float dv = DEL[row * DI_ + d];
        u_v = U[row * DI_ + d];
        if (do_pf) {
          __builtin_prefetch(&DEL[(row + 64) * DI_ + d], 0, 1);
          __builtin_prefetch(&U[(row + 64) * DI_ + d], 0, 1);
        }
        const float du = dv * u_v;
        const float4* Bt4 = (const float4*)&Bs[tt * DS_ + base];
        const float4 bv0 = Bt4[0], bv1 = Bt4[1];
        const float bb[8] = {bv0.x, bv0.y, bv0.z, bv0.w,
                             bv1.x, bv1.y, bv1.z, bv1.w};
        const float r1 = exp2f(g_unit * dv);
        const float r2 = r1 * r1, r3 = r2 * r1, r4 = r2 * r2;
        const float r5 = r4 * r1, r6 = r4 * r2, r7 = r4 * r3, r8 = r4 * r4;
        const float rb = shalf ? r8 : 1.0f;
        const float rr[8] = {r1 * rb, r2 * rb, r3 * rb, r4 * rb,
                             r5 * rb, r6 * rb, r7 * rb, r8 * rb};
#pragma unroll
        for (int j = 0; j < 8; ++j)
          h[j] = fmaf(rr[j], h[j], du * bb[j]);
      }
    } else {
      for (int tt = 0; tt < 64; ++tt) {
        const size_t row = rowbase + t0 + tt;
        const float dv = DEL[row * DI_ + d];
        u_v = U[row * DI_ + d];
        if (do_pf) {
          __builtin_prefetch(&DEL[(row + 64) * DI_ + d], 0, 1);
          __builtin_prefetch(&U[(row + 64) * DI_ + d], 0, 1);
        }
        const float du = dv * u_v;
        const float4* Bt4 = (const float4*)&Bs[tt * DS_ + base];
        const float4 bv0 = Bt4[0], bv1 = Bt4[1];
        const float bb[8] = {bv0.x, bv0.y, bv0.z, bv0.w,
                             bv1.x, bv1.y, bv1.z, bv1.w};
#pragma unroll
        for (int j = 0; j < 8; ++j) {
          float dA = exp2f(dv * as2[j]);
          h[j] = fmaf(dA, h[j], du * bb[j]);
        }
      }
    }
    __syncthreads();
  }

  // y = h . C_last (partial over this thread's 8 states), combine halves.
  float ypart = 0.f;
  const size_t lastrow = rowbase + (L_ - 1);
#pragma unroll
  for (int j = 0; j < 8; ++j)
    ypart = fmaf(h[j], XD[lastrow * NX_ + DTR_ + DS_ + base + j], ypart);
  yred[tid] = ypart;
  __syncthreads();
  if (tid < 128) {
    float y = yred[tid] + yred[tid + 128];
    y = fmaf(u_v, Dp[d], y);
    float zv = fmaf(x[(size_t)b * L_ + (L_ - 1)], v[DI_ + d], c[DI_ + d]);
    YG[(size_t)b * DI_ + d] = y * (zv * fast_sigmoid(zv));
  }
}

// ---------------------------------------------------------------------------
// k5: q[d] = sum_m out_proj_w[d,m] * head_w[m]
// ---------------------------------------------------------------------------
__global__ void k_q(const float* __restrict__ opw, const float* __restrict__ hw,
                    float* __restrict__ Q) {
  __shared__ float red[256];
  int dd = blockIdx.x;
  float a = 0.f;
  for (int m = threadIdx.x; m < DM_; m += 256)
    a = fmaf(opw[(size_t)dd * DM_ + m], hw[m], a);
  red[threadIdx.x] = a;
  __syncthreads();
  for (int off = 128; off > 0; off >>= 1) {
    if (threadIdx.x < off) red[threadIdx.x] += red[threadIdx.x + off];
    __syncthreads();
  }
  if (threadIdx.x == 0) Q[dd] = red[0];
}

// ---------------------------------------------------------------------------
// k6: out[b] = yg[b,:] . q + head_b
// ---------------------------------------------------------------------------
__global__ void k_out(const float* __restrict__ YG, const float* __restrict__ Q,
                      const float* __restrict__ hb, float* __restrict__ out) {
  __shared__ float red[256];
  int b = blockIdx.x;
  float a = 0.f;
  for (int dd = threadIdx.x; dd < DI_; dd += 256)
    a = fmaf(YG[(size_t)b * DI_ + dd], Q[dd], a);
  red[threadIdx.x] = a;
  __syncthreads();
  for (int off = 128; off > 0; off >>= 1) {
    if (threadIdx.x < off) red[threadIdx.x] += red[threadIdx.x + off];
    __syncthreads();
  }
  if (threadIdx.x == 0) out[b] = red[0] + hb[0];
}

// ---------------------------------------------------------------------------
extern "C" void kernel_launch(void* const* d_in, const int* in_sizes, int n_in,
                              void* d_out, int out_size, void* d_ws, size_t ws_size,
                              hipStream_t stream) {
  const float* x         = (const float*)d_in[0];
  const float* w_in      = (const float*)d_in[1];
  const float* b_in      = (const float*)d_in[2];
  const float* in_proj_w = (const float*)d_in[3];
  const float* conv_w    = (const float*)d_in[4];
  const float* conv_b    = (const float*)d_in[5];
  const float* x_proj_w  = (const float*)d_in[6];
  const float* dt_proj_w = (const float*)d_in[7];
  const float* dt_proj_b = (const float*)d_in[8];
  const float* A_log     = (const float*)d_in[9];
  const float* Dp        = (const float*)d_in[10];
  const float* opw       = (const float*)d_in[11];
  const float* hw        = (const float*)d_in[12];
  const float* hb        = (const float*)d_in[13];

  float* ws  = (float*)d_ws;
  float* v   = ws;                                   // 4096
  float* c   = v + 2 * DI_;                          // 4096
  float* U   = c + 2 * DI_;                          // 16384*2048
  float* XD  = U + (size_t)B_ * L_ * DI_;            // 16384*96
  float* DEL = XD + (size_t)B_ * L_ * NX_;           // 16384*2048
  float* YG  = DEL + (size_t)B_ * L_ * DI_;          // 8*2048
  float* Q   = YG + B_ * DI_;                        // 2048
  float* out = (float*)d_out;

  k_vc   <<<(2 * DI_ + 255) / 256, 256, 0, stream>>>(w_in, b_in, in_proj_w, v, c);
  k_uconv<<<(B_ * L_ * DI_ / 4) / 256, 256, 0, stream>>>(
      x, (const float4*)v, (const float4*)c, (const float4*)conv_w,
      (const float4*)conv_b, (float4*)U);
  k_gemmA<<<(B_ * L_ / 16) / 8, 256, 0, stream>>>(U, x_proj_w, XD);
  k_gemmB<<<B_ * L_ / 16, 256, 0, stream>>>(XD, dt_proj_w, dt_proj_b, DEL);
  k_scan <<<B_ * DI_ / 128, 256, 0, stream>>>(DEL, U, XD, A_log, Dp, x, v, c, YG);
  k_q    <<<DI_, 256, 0, stream>>>(opw, hw, Q);
  k_out  <<<B_, 256, 0, stream>>>(YG, Q, hb, out);
}